// ImprovedGATLayer_2637109920386
// MI455X (gfx1250) — compile-verified
//
#include <hip/hip_runtime.h>
#include <math.h>

#define N_NODES   50000
#define IN_F      128
#define H_HEADS   8
#define F_FEAT    16
#define HF        128            // H*F
#define NEG_SLOPE 0.2f
#define BN_EPS    1e-5f

typedef __attribute__((ext_vector_type(2))) float v2f;
typedef __attribute__((ext_vector_type(8))) float v8f;

__device__ __forceinline__ float leaky(float v) { return v >= 0.f ? v : NEG_SLOPE * v; }

// Native non-returning f32 atomics (GLOBAL_ATOMIC_MAX_NUM_F32 / ADD_F32).
__device__ __forceinline__ void atomic_max_f32(float* p, float v) {
#if defined(__HIP_DEVICE_COMPILE__)
  unsafeAtomicMax(p, v);
#else
  atomicMax(p, v);
#endif
}
__device__ __forceinline__ void atomic_add_f32(float* p, float v) {
#if defined(__HIP_DEVICE_COMPILE__)
  unsafeAtomicAdd(p, v);
#else
  atomicAdd(p, v);
#endif
}

// ---------------------------------------------------------------------------
// Kernel 1: h = x @ W  via V_WMMA_F32_16X16X4_F32 (exact fp32 path on CDNA5)
// grid.x = N/16 = 3125 row tiles; block = 256 threads = 8 wave32s.
// wave w computes the 16x16 tile for head w (columns w*16 .. w*16+15).
// A (16x4 f32): lanes 0-15 hold M=0..15 with VGPR0/1 = K=0/1; lanes 16-31 = K=2/3.
// D (16x16 f32): VGPR r: lanes 0-15 -> M=r, lanes 16-31 -> M=r+8, N=lane&15.
// No divergence anywhere -> EXEC all ones, as WMMA requires.
// ---------------------------------------------------------------------------
__global__ void __launch_bounds__(256) gemm_h_wmma(const float* __restrict__ x,
                                                   const float* __restrict__ W,
                                                   float* __restrict__ h) {
  const int lane = threadIdx.x & 31;
  const int wave = threadIdx.x >> 5;           // head / column tile: 0..7
  const int row0 = blockIdx.x * 16;
  const int col0 = wave * 16;
  const int mrow = lane & 15;                  // M (for A) / N (for B,D)
  const int kgrp = (lane >> 4) << 1;           // 0 or 2: which K pair this half-wave holds

  const float* xa = x + (size_t)(row0 + mrow) * IN_F + kgrp;
  const float* wb = W + (size_t)kgrp * HF + col0 + mrow;

  v8f c = {};
  #pragma unroll 4
  for (int k = 0; k < IN_F; k += 4) {
    v2f a, b;
    a.x = xa[k];
    a.y = xa[k + 1];
    b.x = wb[(size_t)k * HF];
    b.y = wb[(size_t)(k + 1) * HF];
    c = __builtin_amdgcn_wmma_f32_16x16x4_f32(
        /*neg_a=*/false, a, /*neg_b=*/false, b,
        /*c_mod=*/(short)0, c, /*reuse_a=*/false, /*reuse_b=*/false);
  }

  const int rbase = row0 + ((lane >> 4) << 3);   // +8 rows for upper half-wave
  #pragma unroll
  for (int r = 0; r < 8; ++r)
    h[(size_t)(rbase + r) * HF + col0 + mrow] = c[r];
}

// ---------------------------------------------------------------------------
// Kernel 2: init m=-inf, denom=0, out=bias (broadcast), BN accumulators = 0
// ---------------------------------------------------------------------------
__global__ void init_state(float* __restrict__ m, float* __restrict__ denom,
                           float* __restrict__ out, const float* __restrict__ bias,
                           float* __restrict__ bnsum, float* __restrict__ bnsumsq,
                           int n) {
  size_t i = (size_t)blockIdx.x * blockDim.x + threadIdx.x;
  size_t total = (size_t)n * HF;
  if (i < total) out[i] = bias[i & (HF - 1)];
  if (i < (size_t)n * H_HEADS) { m[i] = -INFINITY; denom[i] = 0.f; }
  if (i < HF) { bnsum[i] = 0.f; bnsumsq[i] = 0.f; }
}

// ---------------------------------------------------------------------------
// Kernel 3: per-(node,head) attention halves: a_src/a_dst = <h[n,h,:], att>
// thread i == n*8+head, so h offset is exactly i*16 (contiguous b128 loads).
// ---------------------------------------------------------------------------
__global__ void attn_node(const float* __restrict__ h,
                          const float* __restrict__ att_src,
                          const float* __restrict__ att_dst,
                          float* __restrict__ a_src, float* __restrict__ a_dst,
                          int n) {
  int i = blockIdx.x * blockDim.x + threadIdx.x;
  if (i >= n * H_HEADS) return;
  const int head = i & (H_HEADS - 1);
  const float4* hv = (const float4*)(h + (size_t)i * F_FEAT);
  const float4* as = (const float4*)(att_src + head * F_FEAT);
  const float4* ad = (const float4*)(att_dst + head * F_FEAT);
  float s = 0.f, d = 0.f;
  #pragma unroll
  for (int q = 0; q < 4; ++q) {
    float4 v = hv[q], a = as[q], b = ad[q];
    s = fmaf(v.x, a.x, fmaf(v.y, a.y, fmaf(v.z, a.z, fmaf(v.w, a.w, s))));
    d = fmaf(v.x, b.x, fmaf(v.y, b.y, fmaf(v.z, b.z, fmaf(v.w, b.w, d))));
  }
  a_src[i] = s;
  a_dst[i] = d;
}

// ---------------------------------------------------------------------------
// Kernel 4: segment max of raw scores over destination (incl. self loops).
// b128 gathers of the 8 per-head scores; native f32 max atomics.
// ---------------------------------------------------------------------------
__global__ void edge_max(const int* __restrict__ ei,
                         const float* __restrict__ a_src,
                         const float* __restrict__ a_dst,
                         float* __restrict__ m, int E, int n) {
  int e = blockIdx.x * blockDim.x + threadIdx.x;
  int Etot = E + n;
  if (e >= Etot) return;
  int s, d;
  if (e < E) { s = ei[e]; d = ei[E + e]; } else { s = d = e - E; }
  const float4* asv = (const float4*)(a_src + (size_t)s * H_HEADS);
  const float4* adv = (const float4*)(a_dst + (size_t)d * H_HEADS);
  float* mp = m + (size_t)d * H_HEADS;
  #pragma unroll
  for (int q = 0; q < 2; ++q) {
    float4 av = asv[q], bv = adv[q];
    atomic_max_f32(mp + q * 4 + 0, leaky(av.x + bv.x));
    atomic_max_f32(mp + q * 4 + 1, leaky(av.y + bv.y));
    atomic_max_f32(mp + q * 4 + 2, leaky(av.z + bv.z));
    atomic_max_f32(mp + q * 4 + 3, leaky(av.w + bv.w));
  }
}

// ---------------------------------------------------------------------------
// Kernel 5: softmax denominator per destination neighborhood
// ---------------------------------------------------------------------------
__global__ void edge_denom(const int* __restrict__ ei,
                           const float* __restrict__ a_src,
                           const float* __restrict__ a_dst,
                           const float* __restrict__ m,
                           float* __restrict__ denom, int E, int n) {
  int e = blockIdx.x * blockDim.x + threadIdx.x;
  int Etot = E + n;
  if (e >= Etot) return;
  int s, d;
  if (e < E) { s = ei[e]; d = ei[E + e]; } else { s = d = e - E; }
  const float4* asv = (const float4*)(a_src + (size_t)s * H_HEADS);
  const float4* adv = (const float4*)(a_dst + (size_t)d * H_HEADS);
  const float4* mv  = (const float4*)(m + (size_t)d * H_HEADS);
  float* dp = denom + (size_t)d * H_HEADS;
  #pragma unroll
  for (int q = 0; q < 2; ++q) {
    float4 av = asv[q], bv = adv[q], mm = mv[q];
    atomic_add_f32(dp + q * 4 + 0, __expf(leaky(av.x + bv.x) - mm.x));
    atomic_add_f32(dp + q * 4 + 1, __expf(leaky(av.y + bv.y) - mm.y));
    atomic_add_f32(dp + q * 4 + 2, __expf(leaky(av.z + bv.z) - mm.z));
    atomic_add_f32(dp + q * 4 + 3, __expf(leaky(av.w + bv.w) - mm.w));
  }
}

// ---------------------------------------------------------------------------
// Kernel 6: weighted message aggregation.
// One thread per (edge, head): recompute the edge weight (cheap), gather 64B
// of h[src, head, :] as 4x b128, scatter 16 native f32 add atomics.
// ---------------------------------------------------------------------------
__global__ void edge_agg(const int* __restrict__ ei,
                         const float* __restrict__ a_src,
                         const float* __restrict__ a_dst,
                         const float* __restrict__ m,
                         const float* __restrict__ denom,
                         const float* __restrict__ h,
                         float* __restrict__ out, int E, int n) {
  long long t = (long long)blockIdx.x * blockDim.x + threadIdx.x;
  long long Etot = (long long)E + n;
  if (t >= Etot * H_HEADS) return;
  int e = (int)(t >> 3);
  int hh = (int)(t & (H_HEADS - 1));
  int s, d;
  if (e < E) { s = ei[e]; d = ei[E + e]; } else { s = d = e - E; }
  int idx = d * H_HEADS + hh;
  float al = leaky(a_src[s * H_HEADS + hh] + a_dst[idx]);
  float w = __expf(al - m[idx]) / (denom[idx] + 1e-16f);
  const float4* hs = (const float4*)(h + (size_t)s * HF + hh * F_FEAT);
  float* od = out + (size_t)d * HF + hh * F_FEAT;
  #pragma unroll
  for (int q = 0; q < 4; ++q) {
    float4 v = hs[q];
    atomic_add_f32(od + q * 4 + 0, v.x * w);
    atomic_add_f32(od + q * 4 + 1, v.y * w);
    atomic_add_f32(od + q * 4 + 2, v.z * w);
    atomic_add_f32(od + q * 4 + 3, v.w * w);
  }
}

// ---------------------------------------------------------------------------
// Kernel 7: BN column statistics. 128 threads/block (thread = column),
// block-private partial sums over a row stripe, then 2 atomics per column.
// ---------------------------------------------------------------------------
__global__ void __launch_bounds__(HF) bn_stats(const float* __restrict__ out,
                                               float* __restrict__ bnsum,
                                               float* __restrict__ bnsumsq,
                                               int n) {
  int c = threadIdx.x;                          // column 0..127, coalesced reads
  int rows_per_blk = (n + gridDim.x - 1) / gridDim.x;
  int r0 = blockIdx.x * rows_per_blk;
  int r1 = r0 + rows_per_blk; if (r1 > n) r1 = n;
  float s = 0.f, s2 = 0.f;
  for (int r = r0; r < r1; ++r) {
    float v = out[(size_t)r * HF + c];
    s += v;
    s2 = fmaf(v, v, s2);
  }
  atomic_add_f32(&bnsum[c], s);
  atomic_add_f32(&bnsumsq[c], s2);
}

// ---------------------------------------------------------------------------
// Kernel 8: normalize + affine + trailing LeakyReLU (in place on out)
// ---------------------------------------------------------------------------
__global__ void bn_finalize(float* __restrict__ out,
                            const float* __restrict__ bnsum,
                            const float* __restrict__ bnsumsq,
                            const float* __restrict__ gamma,
                            const float* __restrict__ beta, int n) {
  size_t i = (size_t)blockIdx.x * blockDim.x + threadIdx.x;
  if (i >= (size_t)n * HF) return;
  int c = i & (HF - 1);
  float inv_n = 1.f / (float)n;
  float mu = bnsum[c] * inv_n;
  float var = bnsumsq[c] * inv_n - mu * mu;
  float v = (out[i] - mu) * rsqrtf(var + BN_EPS) * gamma[c] + beta[c];
  out[i] = leaky(v);
}

// ---------------------------------------------------------------------------
extern "C" void kernel_launch(void* const* d_in, const int* in_sizes, int n_in,
                              void* d_out, int out_size, void* d_ws, size_t ws_size,
                              hipStream_t stream) {
  const float* x       = (const float*)d_in[0];
  const int*   ei      = (const int*)  d_in[1];
  const float* W       = (const float*)d_in[2];
  const float* att_src = (const float*)d_in[3];
  const float* att_dst = (const float*)d_in[4];
  const float* bias    = (const float*)d_in[5];
  const float* gamma   = (const float*)d_in[6];
  const float* beta    = (const float*)d_in[7];
  float* out = (float*)d_out;

  const int n = in_sizes[0] / IN_F;     // 50000
  const int E = in_sizes[1] / 2;        // 800000
  const int Etot = E + n;

  // workspace carve-up (floats), ~32 MB total
  float* ws      = (float*)d_ws;
  float* h       = ws;                            // n*128
  float* a_src   = h + (size_t)n * HF;            // n*8
  float* a_dst   = a_src + (size_t)n * H_HEADS;   // n*8
  float* m       = a_dst + (size_t)n * H_HEADS;   // n*8
  float* denom   = m + (size_t)n * H_HEADS;       // n*8
  float* bnsum   = denom + (size_t)n * H_HEADS;   // 128
  float* bnsumsq = bnsum + HF;                    // 128

  // 1) projection via WMMA (n is a multiple of 16: 50000 = 3125*16)
  gemm_h_wmma<<<n / 16, 256, 0, stream>>>(x, W, h);

  // 2) init accumulators / output=bias
  {
    size_t total = (size_t)n * HF;
    int blocks = (int)((total + 255) / 256);
    init_state<<<blocks, 256, 0, stream>>>(m, denom, out, bias, bnsum, bnsumsq, n);
  }

  // 3) per-(node,head) attention halves
  attn_node<<<(n * H_HEADS + 255) / 256, 256, 0, stream>>>(h, att_src, att_dst,
                                                           a_src, a_dst, n);

  // 4) segment max
  edge_max<<<(Etot + 255) / 256, 256, 0, stream>>>(ei, a_src, a_dst, m, E, n);

  // 5) softmax denominator
  edge_denom<<<(Etot + 255) / 256, 256, 0, stream>>>(ei, a_src, a_dst, m, denom, E, n);

  // 6) weighted aggregation
  {
    long long work = (long long)Etot * H_HEADS;
    int blocks = (int)((work + 255) / 256);
    edge_agg<<<blocks, 256, 0, stream>>>(ei, a_src, a_dst, m, denom, h, out, E, n);
  }

  // 7) BN stats + 8) finalize
  bn_stats<<<256, HF, 0, stream>>>(out, bnsum, bnsumsq, n);
  {
    size_t total = (size_t)n * HF;
    int blocks = (int)((total + 255) / 256);
    bn_finalize<<<blocks, 256, 0, stream>>>(out, bnsum, bnsumsq, gamma, beta, n);
  }
}